// QuantumRegressionModel_65481071397597
// MI455X (gfx1250) — compile-verified
//
#include <hip/hip_runtime.h>

#define NW      12
#define DIM     4096      // 2^12
#define NLAYERS 3
#define TPB     256       // 8 wave32s

typedef __attribute__((ext_vector_type(2))) float v2f;
typedef __attribute__((ext_vector_type(8))) float v8f;

__device__ __forceinline__ float2 cmul(float2 a, float2 b) {
    return make_float2(a.x * b.x - a.y * b.y, a.x * b.y + a.y * b.x);
}

__global__ __launch_bounds__(TPB) void quantum_circuit_fused(
    const float* __restrict__ state,     // (B, DIM) real initial amplitudes
    const float* __restrict__ weights,   // (3, 12, 3)
    const float* __restrict__ head_w,    // (1, 12)
    const float* __restrict__ head_b,    // (1,)
    float* __restrict__ out)             // (B,)
{
    __shared__ float2 S[DIM];                 // 32 KB: complex state
    __shared__ float2 Ush[256];               // fused 16x16 group unitary
    __shared__ float2 u2[NLAYERS * NW * 4];   // 36 per-wire 2x2 gates
    __shared__ float  red[TPB];

    const int b    = blockIdx.x;
    const int tid  = threadIdx.x;
    const int lane = tid & 31;
    const int wave = tid >> 5;

    // ---- Phase 0: build all 2x2 gates u = RZ(c) @ RY(b) @ RX(a) ----
    if (tid < NLAYERS * NW) {
        const float a  = weights[tid * 3 + 0];
        const float bb = weights[tid * 3 + 1];
        const float c  = weights[tid * 3 + 2];
        const float ca = cosf(0.5f * a),  sa = sinf(0.5f * a);
        const float cb = cosf(0.5f * bb), sb = sinf(0.5f * bb);
        const float cc = cosf(0.5f * c),  sc = sinf(0.5f * c);
        // M = RY @ RX
        const float2 m00 = make_float2( cb * ca,  sb * sa);
        const float2 m01 = make_float2(-sb * ca, -cb * sa);
        const float2 m10 = make_float2( sb * ca, -cb * sa);
        const float2 m11 = make_float2( cb * ca, -sb * sa);
        const float2 e   = make_float2(cc, -sc);   // e^{-i c/2}
        const float2 ec  = make_float2(cc,  sc);   // e^{+i c/2}
        u2[tid * 4 + 0] = cmul(e,  m00);
        u2[tid * 4 + 1] = cmul(e,  m01);
        u2[tid * 4 + 2] = cmul(ec, m10);
        u2[tid * 4 + 3] = cmul(ec, m11);
    }

    // ---- Load initial state (real -> complex) ----
    #pragma unroll
    for (int i = 0; i < DIM / TPB; ++i) {
        const int j = tid + i * TPB;
        S[j] = make_float2(state[(size_t)b * DIM + j], 0.0f);
    }

    const int m  = lane & 15;          // A-matrix row / B-matrix column select
    const int kb = (lane >> 4) << 1;   // K sub-offset within a k-chunk

    for (int l = 0; l < NLAYERS; ++l) {
        for (int g = 0; g < 3; ++g) {
            __syncthreads();   // state coherent; Ush free for reuse

            // ---- Build fused 16x16 unitary for wires 4g..4g+3 (kron) ----
            {
                const int i = tid >> 4, j = tid & 15;
                const int wbase = (l * NW + g * 4) * 4;
                float2 p = u2[wbase +  0 + ((i >> 3) & 1) * 2 + ((j >> 3) & 1)];
                p = cmul(p, u2[wbase +  4 + ((i >> 2) & 1) * 2 + ((j >> 2) & 1)]);
                p = cmul(p, u2[wbase +  8 + ((i >> 1) & 1) * 2 + ((j >> 1) & 1)]);
                p = cmul(p, u2[wbase + 12 + ( i       & 1) * 2 + ( j       & 1)]);
                Ush[tid] = p;
            }
            __syncthreads();

            const int rs = (g == 0) ? 256 : (g == 1) ? 16 : 1;

            // ---- A fragments: U rows (16x4 f32 WMMA layout) ----
            v2f ar[4], ai[4], ain[4];
            #pragma unroll
            for (int kc = 0; kc < 4; ++kc) {
                const float2 e0 = Ush[m * 16 + kc * 4 + kb];
                const float2 e1 = Ush[m * 16 + kc * 4 + kb + 1];
                ar[kc]  = (v2f){ e0.x,  e1.x};
                ai[kc]  = (v2f){ e0.y,  e1.y};
                ain[kc] = (v2f){-e0.y, -e1.y};
            }

            // ---- Each wave owns 2 disjoint 16-column tiles: in-place GEMM ----
            #pragma unroll
            for (int t = 0; t < 2; ++t) {
                const int n  = ((wave << 1) + t) * 16 + m;   // column index 0..255
                const int ba = (g == 0) ? n
                             : (g == 1) ? ((n >> 4) * 256 + (n & 15))
                                        : (n * 16);

                v2f br[4], bi[4];
                #pragma unroll
                for (int kc = 0; kc < 4; ++kc) {
                    const int k0 = kc * 4 + kb;
                    const float2 s0 = S[ba + k0 * rs];
                    const float2 s1 = S[ba + (k0 + 1) * rs];
                    br[kc] = (v2f){s0.x, s1.x};
                    bi[kc] = (v2f){s0.y, s1.y};
                }

                v8f cr = {}; v8f ci = {};
                #pragma unroll
                for (int kc = 0; kc < 4; ++kc) {
                    // Cr += Ur*Sr - Ui*Si ; Ci += Ur*Si + Ui*Sr
                    cr = __builtin_amdgcn_wmma_f32_16x16x4_f32(false, ar[kc],  false, br[kc], (short)0, cr, false, false);
                    cr = __builtin_amdgcn_wmma_f32_16x16x4_f32(false, ain[kc], false, bi[kc], (short)0, cr, false, false);
                    ci = __builtin_amdgcn_wmma_f32_16x16x4_f32(false, ar[kc],  false, bi[kc], (short)0, ci, false, false);
                    ci = __builtin_amdgcn_wmma_f32_16x16x4_f32(false, ai[kc],  false, br[kc], (short)0, ci, false, false);
                }

                const int Mb = (lane >> 4) * 8;
                #pragma unroll
                for (int v = 0; v < 8; ++v)
                    S[ba + (Mb + v) * rs] = make_float2(cr[v], ci[v]);
            }
        }

        // ---- CNOT chain == Gray-code permutation: psi'[j] = psi[j ^ (j>>1)] ----
        __syncthreads();
        float2 tmp[DIM / TPB];
        #pragma unroll
        for (int i = 0; i < DIM / TPB; ++i) {
            const int j = tid + i * TPB;
            tmp[i] = S[j ^ (j >> 1)];
        }
        __syncthreads();
        #pragma unroll
        for (int i = 0; i < DIM / TPB; ++i)
            S[tid + i * TPB] = tmp[i];
    }
    __syncthreads();

    // ---- Measurement: out[b] = sum_j |psi_j|^2 * sum_w hw[w]*(1-2*bit_{11-w}(j)) + hb ----
    float hw[NW];
    #pragma unroll
    for (int w = 0; w < NW; ++w) hw[w] = head_w[w];

    float acc = 0.0f;
    #pragma unroll
    for (int i = 0; i < DIM / TPB; ++i) {
        const int j = tid + i * TPB;
        const float2 p = S[j];
        const float pr = p.x * p.x + p.y * p.y;
        float s = 0.0f;
        #pragma unroll
        for (int w = 0; w < NW; ++w)
            s += ((j >> (11 - w)) & 1) ? -hw[w] : hw[w];
        acc += pr * s;
    }

    red[tid] = acc;
    __syncthreads();
    #pragma unroll
    for (int off = TPB / 2; off > 0; off >>= 1) {
        if (tid < off) red[tid] += red[tid + off];
        __syncthreads();
    }
    if (tid == 0) out[b] = red[0] + head_b[0];
}

extern "C" void kernel_launch(void* const* d_in, const int* in_sizes, int n_in,
                              void* d_out, int out_size, void* d_ws, size_t ws_size,
                              hipStream_t stream) {
    const float* state   = (const float*)d_in[0];  // (512, 4096) f32
    const float* weights = (const float*)d_in[1];  // (3, 12, 3)  f32
    const float* head_w  = (const float*)d_in[2];  // (1, 12)     f32
    const float* head_b  = (const float*)d_in[3];  // (1,)        f32
    float* out = (float*)d_out;                    // (512,)      f32

    const int batch = out_size;                    // 512, one block per state
    quantum_circuit_fused<<<dim3(batch), dim3(TPB), 0, stream>>>(
        state, weights, head_w, head_b, out);
}